// CoPE2d_v2_40252433498182
// MI455X (gfx1250) — compile-verified
//
#include <hip/hip_runtime.h>

typedef __attribute__((ext_vector_type(2))) float v2f;
typedef __attribute__((ext_vector_type(8))) float v8f;

#define HEADS 12
#define SIDE 24
#define SEQ 576
#define HD 64
#define NPOS 64
#define GB 96            // B * HEADS
#define OUTD 577
#define OUTSLICE (577 * 577)

// ---------------------------------------------------------------------------
// Kernel 1: CoPE unit. One block per (direction, bh, head).
// Produces w_x / w_y in final [96, 576, 24] layout + per-row squared norms.
// ---------------------------------------------------------------------------
__global__ __launch_bounds__(256)
void cope_unit_kernel(const float* __restrict__ q, const float* __restrict__ k,
                      const float* __restrict__ attn,
                      const float* __restrict__ pe_x, const float* __restrict__ pe_y,
                      const int* __restrict__ is_cope_k,
                      float* __restrict__ wx, float* __restrict__ wy,
                      float* __restrict__ sqx, float* __restrict__ sqy)
{
    __shared__ float lPos[SIDE][SIDE + 1];   // gates, then clipped reverse-cumsum
    __shared__ float lSrc[SIDE][HD];         // 24 rows of q/k
    __shared__ float lPE[HD][NPOS];          // pos_emb [d][n]
    __shared__ float lLI[SIDE][NPOS];        // logits_int
    __shared__ float lW[SIDE][SIDE + 1];     // interpolated output tile

    const int tid = threadIdx.x;
    int bi = blockIdx.x;
    const int dir = bi / (192 * HEADS);      // 0 = x, 1 = y
    bi -= dir * (192 * HEADS);
    const int bh   = bi / HEADS;             // 0..191
    const int head = bi - bh * HEADS;        // 0..11
    const int b = bh / SIDE;
    const int s = bh - b * SIDE;             // h_idx (x-dir) or w_idx (y-dir)
    const int g = b * HEADS + head;          // flat batch-head 0..95

    const float* src = (is_cope_k[0] != 0) ? q : k;   // is_cope_k -> src = q
    const float* pe  = dir ? pe_y : pe_x;

    // gates = sigmoid(attn[bh, head])
    const float* attn_tile = attn + (size_t)(bh * HEADS + head) * (SIDE * SIDE);
    for (int t = tid; t < SIDE * SIDE; t += 256) {
        int r = t / SIDE, c = t - r * SIDE;
        lPos[r][c] = 1.0f / (1.0f + __expf(-attn_tile[t]));
    }
    // src rows: x-dir seqpos = s*24 + r ; y-dir seqpos = r*24 + s
    for (int t = tid; t < SIDE * HD; t += 256) {
        int r = t / HD, d = t - r * HD;
        int seqpos = dir ? (r * SIDE + s) : (s * SIDE + r);
        lSrc[r][d] = src[((size_t)g * SEQ + seqpos) * HD + d];
    }
    for (int t = tid; t < HD * NPOS; t += 256)
        ((float*)lPE)[t] = pe[t];
    __syncthreads();

    // reversed cumulative sum, clipped at NPOS-1
    if (tid < SIDE) {
        float run = 0.0f;
        for (int c = SIDE - 1; c >= 0; --c) {
            run += lPos[tid][c];
            lPos[tid][c] = fminf(run, (float)(NPOS - 1));
        }
    }
    // logits_int[r][n] = sum_d src[r][d] * pe[d][n]
    for (int t = tid; t < SIDE * NPOS; t += 256) {
        int r = t / NPOS, n = t - r * NPOS;
        float acc = 0.0f;
#pragma unroll 8
        for (int d = 0; d < HD; ++d)
            acc = fmaf(lSrc[r][d], lPE[d][n], acc);
        lLI[r][n] = acc;
    }
    __syncthreads();

    // gather + lerp, store in final transposed layout
    float* wout = dir ? wy : wx;
    for (int t = tid; t < SIDE * SIDE; t += 256) {
        int r = t / SIDE, c = t - r * SIDE;
        float pos = lPos[r][c];
        float pf  = floorf(pos);
        int fi = (int)pf;
        int ci = (int)ceilf(pos);
        fi = fi < 0 ? 0 : (fi > NPOS - 1 ? NPOS - 1 : fi);
        ci = ci < 0 ? 0 : (ci > NPOS - 1 ? NPOS - 1 : ci);
        float wgt = pos - pf;
        float val = lLI[r][ci] * wgt + lLI[r][fi] * (1.0f - wgt);
        lW[r][c] = val;
        int row = dir ? (r * SIDE + s) : (s * SIDE + r);
        wout[((size_t)g * SEQ + row) * SIDE + c] = val;
    }
    __syncthreads();

    // per-row squared norms for cdist
    if (tid < SIDE) {
        float acc = 0.0f;
        for (int c = 0; c < SIDE; ++c)
            acc = fmaf(lW[tid][c], lW[tid][c], acc);
        int row = dir ? (tid * SIDE + s) : (s * SIDE + tid);
        (dir ? sqy : sqx)[(size_t)g * SEQ + row] = acc;
    }
}

// ---------------------------------------------------------------------------
// Kernel 2: cdist via V_WMMA_F32_16X16X4_F32 Gram tiles, x + y fused,
// masked sqrt, write into padded [577,577] output slice.
// Block = (g, ti); 8 waves each sweep column tiles tj = wave, wave+8, ...
// ---------------------------------------------------------------------------
__global__ __launch_bounds__(256)
void cdist_wmma_kernel(const float* __restrict__ wx, const float* __restrict__ wy,
                       const float* __restrict__ sqx, const float* __restrict__ sqy,
                       float* __restrict__ out)
{
    const int g  = blockIdx.x / 36;
    const int ti = blockIdx.x - g * 36;
    const int lane = threadIdx.x & 31;
    const int wave = threadIdx.x >> 5;
    const int half = lane >> 4;          // 0: lanes 0-15, 1: lanes 16-31
    const int l15  = lane & 15;

    const float* Xx = wx + (size_t)g * SEQ * SIDE;
    const float* Xy = wy + (size_t)g * SEQ * SIDE;

    // A fragment (fixed for this block): lane holds row M = ti*16 + l15,
    // K pair (half*2, half*2+1) within each K=4 step.
    const int rowA = ti * 16 + l15;
    const int koff = half * 2;
    v2f ax[6], ay[6];
#pragma unroll
    for (int kk = 0; kk < 6; ++kk) {
        int o = rowA * SIDE + kk * 4 + koff;
        ax[kk].x = Xx[o]; ax[kk].y = Xx[o + 1];
        ay[kk].x = Xy[o]; ay[kk].y = Xy[o + 1];
    }
    // row norms for the 8 M values this lane owns in C/D
    float six[8], siy[8];
#pragma unroll
    for (int r = 0; r < 8; ++r) {
        int i = ti * 16 + half * 8 + r;
        six[r] = sqx[(size_t)g * SEQ + i];
        siy[r] = sqy[(size_t)g * SEQ + i];
    }

    for (int tj = wave; tj < 36; tj += 8) {
        // B fragment: B[k,j] = X[j,k] -> same load pattern as A with rows tj*16..
        const int rowB = tj * 16 + l15;
        v2f bx[6], by[6];
#pragma unroll
        for (int kk = 0; kk < 6; ++kk) {
            int o = rowB * SIDE + kk * 4 + koff;
            bx[kk].x = Xx[o]; bx[kk].y = Xx[o + 1];
            by[kk].x = Xy[o]; by[kk].y = Xy[o + 1];
        }
        float sjx = sqx[(size_t)g * SEQ + tj * 16 + l15];
        float sjy = sqy[(size_t)g * SEQ + tj * 16 + l15];

        v8f cx = {};
        v8f cy = {};
#pragma unroll
        for (int kk = 0; kk < 6; ++kk) {
            cx = __builtin_amdgcn_wmma_f32_16x16x4_f32(
                false, ax[kk], false, bx[kk], (short)0, cx, false, false);
            cy = __builtin_amdgcn_wmma_f32_16x16x4_f32(
                false, ay[kk], false, by[kk], (short)0, cy, false, false);
        }

#pragma unroll
        for (int r = 0; r < 8; ++r) {
            int i = ti * 16 + half * 8 + r;
            int j = tj * 16 + l15;
            float d2x = fmaxf(six[r] + sjx - 2.0f * cx[r], 0.0f);
            float vx = (d2x > 0.0f) ? sqrtf(d2x) : 0.0f;
            float d2y = fmaxf(siy[r] + sjy - 2.0f * cy[r], 0.0f);
            float vy = (d2y > 0.0f) ? sqrtf(d2y) : 0.0f;
            out[(size_t)g * OUTSLICE + (size_t)(i + 1) * OUTD + (j + 1)] = vx + vy;
        }
    }
}

// ---------------------------------------------------------------------------
// Kernel 3: zero the pad row 0 / col 0 of each [577,577] slice.
// ---------------------------------------------------------------------------
__global__ void pad_zero_kernel(float* __restrict__ out)
{
    int t = blockIdx.x * blockDim.x + threadIdx.x;
    if (t >= GB * OUTD) return;
    int g = t / OUTD, p = t - g * OUTD;
    out[(size_t)g * OUTSLICE + p] = 0.0f;                 // row 0
    out[(size_t)g * OUTSLICE + (size_t)p * OUTD] = 0.0f;  // col 0
}

extern "C" void kernel_launch(void* const* d_in, const int* in_sizes, int n_in,
                              void* d_out, int out_size, void* d_ws, size_t ws_size,
                              hipStream_t stream)
{
    const float* q    = (const float*)d_in[0];
    const float* k    = (const float*)d_in[1];
    /* d_in[2] = value: unused by the reference */
    const float* attn = (const float*)d_in[3];
    const float* pe_x = (const float*)d_in[4];
    const float* pe_y = (const float*)d_in[5];
    const int* is_cope_k = (const int*)d_in[6];

    float* wx  = (float*)d_ws;                       // [96, 576, 24]
    float* wy  = wx + (size_t)GB * SEQ * SIDE;       // [96, 576, 24]
    float* sqx = wy + (size_t)GB * SEQ * SIDE;       // [96, 576]
    float* sqy = sqx + (size_t)GB * SEQ;             // [96, 576]
    float* out = (float*)d_out;                      // [96, 577, 577]

    cope_unit_kernel<<<2 * 192 * HEADS, 256, 0, stream>>>(
        q, k, attn, pe_x, pe_y, is_cope_k, wx, wy, sqx, sqy);

    cdist_wmma_kernel<<<GB * 36, 256, 0, stream>>>(wx, wy, sqx, sqy, out);

    pad_zero_kernel<<<(GB * OUTD + 255) / 256, 256, 0, stream>>>(out);
}